// QuantumGate_65481071406811
// MI455X (gfx1250) — compile-verified
//
#include <hip/hip_runtime.h>

// QuantumGate closed form for MI455X (gfx1250):
//   z[b,j] = prod_{i in S_j} cos(x[b,i])
// RZ phases cancel in |psi|^2; the CNOT chain is the GF(2) prefix-XOR matrix L;
// after DEPTH layers output bit j = XOR of input bits S_j = row j of L^DEPTH.
// (L^d)_{j,i} = C((j-i)+(d-1), d-1) mod 2 = 1 iff i<=j && ((d-1)&(j-i))==0 (Lucas).
// Expectation factorizes over independent RX wires -> params provably drop out.
//
// Traffic is 4096x12 f32 in + out (~393 KB): pure launch-latency bound at 23.3 TB/s.
// Math done as two register-resident GEMMs on V_WMMA_F32_16X16X4_F32 (K=12, 3 chunks):
//   A (16x4) = 0/1 subset matrix (M = wire j), shared by both GEMMs
//   B (4x16) = log2|cos| values / negative-sign flags (N = batch row)
//   D_mag[j][row] = sum of logs -> exp2;  D_par[j][row] = #negatives -> sign parity
// D layout gives each lane contiguous wires -> 16B-aligned global_store_b128.

typedef float v2f __attribute__((ext_vector_type(2)));
typedef float v4f __attribute__((ext_vector_type(4)));
typedef float v8f __attribute__((ext_vector_type(8)));

#define NQ 12

__global__ __launch_bounds__(256) void qgate_wmma_kernel(
    const float* __restrict__ x, float* __restrict__ out, int batch, int dm1)
{
    const int lane = threadIdx.x & 31;
    const int wave = (int)((blockIdx.x * blockDim.x + threadIdx.x) >> 5);
    const int row0 = wave << 4;                 // 16 batch rows per wave
    if (row0 >= batch) return;                  // wave-uniform: EXEC stays all-ones

    const int hi = lane >> 4;                   // half-wave (K offset +2 per ISA layout)
    const int n  = lane & 15;                   // lane's N column (batch row) and A M-row (wire)
    int lrow = row0 + n;
    if (lrow > batch - 1) lrow = batch - 1;     // clamp loads for partial tiles
    const float* xr = x + (size_t)lrow * NQ;    // rows are 48B -> 16B aligned

    float xv[NQ];
    *(v4f*)(xv + 0) = *(const v4f*)(xr + 0);
    *(v4f*)(xv + 4) = *(const v4f*)(xr + 4);
    *(v4f*)(xv + 8) = *(const v4f*)(xr + 8);

    float lg[NQ], nf[NQ];
    #pragma unroll
    for (int i = 0; i < NQ; ++i) {
        float c = __cosf(xv[i]);                // native v_cos_f32; |x| ~ N(0,1)
        nf[i] = (c < 0.0f) ? 1.0f : 0.0f;       // sign flag as exact small float
        lg[i] = fmaxf(__log2f(fabsf(c)), -126.0f);  // clamp: no 0*(-inf) in WMMA
    }

    // Two GEMMs sharing A: D = A(16x4) * B(4x16) + C over 3 K-chunks.
    // Lane's K pair: 4c + 2*hi (+0/+1). Membership via Lucas' theorem.
    v8f accL = {0.f,0.f,0.f,0.f,0.f,0.f,0.f,0.f};   // sum of log2|cos|
    v8f accP = {0.f,0.f,0.f,0.f,0.f,0.f,0.f,0.f};   // count of negative factors
    #pragma unroll
    for (int c = 0; c < 3; ++c) {
        const int k0 = 4 * c;
        const int kA = k0 + 2 * hi;
        v2f A, Bl, Bp;
        A.x  = ((kA     <= n) && ((dm1 & (n - kA    )) == 0)) ? 1.0f : 0.0f;
        A.y  = ((kA + 1 <= n) && ((dm1 & (n - kA - 1)) == 0)) ? 1.0f : 0.0f;
        Bl.x = hi ? lg[k0 + 2] : lg[k0 + 0];    // constant indices -> single cndmask
        Bl.y = hi ? lg[k0 + 3] : lg[k0 + 1];
        Bp.x = hi ? nf[k0 + 2] : nf[k0 + 0];
        Bp.y = hi ? nf[k0 + 3] : nf[k0 + 1];
        accL = __builtin_amdgcn_wmma_f32_16x16x4_f32(
            false, A, false, Bl, (short)0, accL, false, false);
        accP = __builtin_amdgcn_wmma_f32_16x16x4_f32(
            false, A, false, Bp, (short)0, accP, false, false);
    }

    // D layout: VGPR r = wire (r + 8*hi) of batch row n. Contiguous 4-wide stores.
    const int row = row0 + n;
    if (row < batch) {
        float* o = out + (size_t)row * NQ + 8 * hi;   // byte offs 48*row + 32*hi: 16B aligned
        v4f z0;
        #pragma unroll
        for (int r = 0; r < 4; ++r) {
            float mag = exp2f(accL[r]);               // v_exp_f32 + ldexp fixup
            z0[r] = (((int)accP[r]) & 1) ? -mag : mag; // counts <= 12: exact in f32
        }
        *(v4f*)o = z0;                                // wires 8*hi .. 8*hi+3
        if (hi == 0) {                                // wires 4..7 (hi=1's 12..15 are padding)
            v4f z1;
            #pragma unroll
            for (int r = 4; r < 8; ++r) {
                float mag = exp2f(accL[r]);
                z1[r - 4] = (((int)accP[r]) & 1) ? -mag : mag;
            }
            *(v4f*)(o + 4) = z1;
        }
    }
}

extern "C" void kernel_launch(void* const* d_in, const int* in_sizes, int n_in,
                              void* d_out, int out_size, void* d_ws, size_t ws_size,
                              hipStream_t stream)
{
    const float* x = (const float*)d_in[0];
    float* out = (float*)d_out;
    const int batch = in_sizes[0] / NQ;
    const int depth = (n_in > 1) ? (in_sizes[1] / NQ) : 0;   // params values provably unused

    const int waves   = (batch + 15) / 16;
    const int threads = 256;                                  // 8 waves / block
    const int blocks  = (waves * 32 + threads - 1) / threads;
    qgate_wmma_kernel<<<blocks, threads, 0, stream>>>(x, out, batch, depth - 1);
}